// GNN_82205674045701
// MI455X (gfx1250) — compile-verified
//
#include <hip/hip_runtime.h>

#define D_FEAT 64
#define D_OUT 128

// ---------------------------------------------------------------------------
// Kernel: edge scatter (placed FIRST in the file so the disasm snippet shows
// whether unsafeAtomicAdd lowered to hardware global_atomic_add_f32).
//
// 16 lanes cooperate per edge; each lane moves one float4 of x[row[e]] into
// out[col[e], 0:64] via L2-side f32 atomics (results unused -> non-returning,
// STOREcnt path). Lane 0 counts the edge with an integer atomic
// (deterministic). Edge-index loads are non-temporal: es is streamed once
// (8 MB, zero reuse) while x/out are the reused L2-resident working set.
// ---------------------------------------------------------------------------
__global__ void gnn_scatter_kernel(const float* __restrict__ x,
                                   const int* __restrict__ col,
                                   const int* __restrict__ row,
                                   float* __restrict__ out,
                                   unsigned* __restrict__ count,
                                   unsigned n_edges) {
    unsigned tid = blockIdx.x * blockDim.x + threadIdx.x;
    unsigned e = tid >> 4;   // 16 lanes per edge
    unsigned l = tid & 15u;  // lane-in-edge: 4 floats each
    if (e >= n_edges) return;

    // Streaming, zero-reuse index loads -> non-temporal hint.
    unsigned c = (unsigned)__builtin_nontemporal_load(col + e);
    unsigned r = (unsigned)__builtin_nontemporal_load(row + e);

    // Coalesced 256B gather per 16-lane group (GLOBAL_LOAD_B128);
    // x is 25.6 MB -> L2-resident on the 192 MB global L2.
    const float4 v = *reinterpret_cast<const float4*>(
        x + r * (unsigned)D_FEAT + (l << 2));

    float* dst = out + c * (unsigned)D_OUT + (l << 2);
    // unsafeAtomicAdd guarantees hardware global_atomic_add_f32 (no CAS loop);
    // return value unused -> non-returning atomic.
    unsafeAtomicAdd(dst + 0, v.x);
    unsafeAtomicAdd(dst + 1, v.y);
    unsafeAtomicAdd(dst + 2, v.z);
    unsafeAtomicAdd(dst + 3, v.w);

    if (l == 0u) {
        atomicAdd(count + c, 1u);  // global_atomic_add_u32, deterministic
    }
}

// ---------------------------------------------------------------------------
// Kernel: zero the accumulated half of out (out[:, 0:64]) and the per-node
// edge counter in workspace. Must run every call (ws is not re-poisoned).
// ---------------------------------------------------------------------------
__global__ void gnn_init_kernel(float* __restrict__ out,
                                unsigned* __restrict__ count,
                                unsigned n_nodes) {
    unsigned tid = blockIdx.x * blockDim.x + threadIdx.x;
    unsigned total = n_nodes * (unsigned)D_FEAT;
    if (tid < total) {
        unsigned node = tid >> 6;   // tid / 64
        unsigned d    = tid & 63u;  // tid % 64
        out[node * (unsigned)D_OUT + d] = 0.0f;
    }
    if (tid < n_nodes) {
        count[tid] = 0u;
    }
}

// ---------------------------------------------------------------------------
// Kernel: finalize. Per node: first half /= max(count,1); second half is
// analytically count>0 ? x[n] : 0 (the x[col] half of the scatter-mean
// collapses to this), so no atomics were ever needed for it.
// ---------------------------------------------------------------------------
__global__ void gnn_finalize_kernel(const float* __restrict__ x,
                                    const unsigned* __restrict__ count,
                                    float* __restrict__ out,
                                    unsigned n_nodes) {
    unsigned tid = blockIdx.x * blockDim.x + threadIdx.x;
    unsigned node = tid >> 4;   // 16 lanes per node
    unsigned l    = tid & 15u;
    if (node >= n_nodes) return;

    unsigned cnt = count[node];
    float inv = 1.0f / (float)(cnt > 0u ? cnt : 1u);

    float4* o_lo = reinterpret_cast<float4*>(
        out + node * (unsigned)D_OUT + (l << 2));
    float4 s = *o_lo;
    s.x *= inv; s.y *= inv; s.z *= inv; s.w *= inv;
    *o_lo = s;

    float4 xv = *reinterpret_cast<const float4*>(
        x + node * (unsigned)D_FEAT + (l << 2));
    if (cnt == 0u) { xv.x = 0.0f; xv.y = 0.0f; xv.z = 0.0f; xv.w = 0.0f; }
    *reinterpret_cast<float4*>(
        out + node * (unsigned)D_OUT + D_FEAT + (l << 2)) = xv;
}

// ---------------------------------------------------------------------------
// Launch wrapper
// ---------------------------------------------------------------------------
extern "C" void kernel_launch(void* const* d_in, const int* in_sizes, int n_in,
                              void* d_out, int out_size, void* d_ws, size_t ws_size,
                              hipStream_t stream) {
    const float* x  = (const float*)d_in[0];          // [N, 64]
    const int*   es = (const int*)d_in[1];            // [2, E]
    float* out = (float*)d_out;                       // [N, 128]

    const unsigned n_nodes = (unsigned)(in_sizes[0] / D_FEAT);
    const unsigned n_edges = (unsigned)(in_sizes[1] / 2);

    const int* col = es;            // es[0]
    const int* row = es + n_edges;  // es[1]

    unsigned* count = (unsigned*)d_ws;  // N * 4 bytes of scratch

    const int block = 256;  // 8 wave32 waves per workgroup

    // 1) init: n_nodes*64 threads (first n_nodes also zero the counters)
    {
        unsigned total = n_nodes * D_FEAT;
        unsigned grid = (total + block - 1) / block;
        gnn_init_kernel<<<grid, block, 0, stream>>>(out, count, n_nodes);
    }

    // 2) scatter: 16 threads per edge
    {
        unsigned total = n_edges * 16u;
        unsigned grid = (total + block - 1) / block;
        gnn_scatter_kernel<<<grid, block, 0, stream>>>(x, col, row, out, count,
                                                       n_edges);
    }

    // 3) finalize: 16 threads per node
    {
        unsigned total = n_nodes * 16u;
        unsigned grid = (total + block - 1) / block;
        gnn_finalize_kernel<<<grid, block, 0, stream>>>(x, count, out, n_nodes);
    }
}